// LightGCN_26371099197484
// MI455X (gfx1250) — compile-verified
//
#include <hip/hip_runtime.h>
#include <hip/hip_bf16.h>

typedef __attribute__((ext_vector_type(2))) float v2f;
typedef __attribute__((ext_vector_type(8))) float v8f;

#define DFEAT 64

// ---------------- elementwise helpers ----------------

__global__ void k_init(const float* __restrict__ uf, const float* __restrict__ itf,
                       float* __restrict__ acc, float* __restrict__ f0,
                       float* __restrict__ f1, long n, long nu_elems) {
  long i = (long)blockIdx.x * blockDim.x + threadIdx.x;
  if (i >= n) return;
  float v = (i < nu_elems) ? uf[i] : itf[i - nu_elems];
  acc[i] = v;
  f0[i] = v;
  f1[i] = 0.0f;
}

__global__ void k_acc_add_zero(float* __restrict__ acc, const float* __restrict__ f,
                               float* __restrict__ fz, long n) {
  long i = (long)blockIdx.x * blockDim.x + threadIdx.x;
  if (i >= n) return;
  acc[i] += f[i];
  fz[i] = 0.0f;
}

__global__ void k_acc_add_scale(float* __restrict__ acc, const float* __restrict__ f, long n) {
  long i = (long)blockIdx.x * blockDim.x + threadIdx.x;
  if (i >= n) return;
  acc[i] = (acc[i] + f[i]) * (1.0f / 3.0f);
}

__global__ void k_zero(float* __restrict__ p, long n) {
  long i = (long)blockIdx.x * blockDim.x + threadIdx.x;
  if (i >= n) return;
  p[i] = 0.0f;
}

// ---------------- edge-parallel SpMM: wave per edge ----------------
// 32 lanes per edge; lane reads float2 (8B) of source row -> 256B coalesced,
// scatters with hardware f32 atomics (output rows are L2-resident).

__global__ void k_spmm(const int* __restrict__ rows, const int* __restrict__ cols,
                       const float* __restrict__ vals, const float* __restrict__ fin,
                       float* __restrict__ fout, int nnz, int coloff) {
  long gid = (long)blockIdx.x * blockDim.x + threadIdx.x;
  int e = (int)(gid >> 5);
  int lane = (int)(gid & 31);
  if (e >= nnz) return;
  int r = rows[e];
  int c = cols[e];
  float v = vals[e];
  const float2* src = (const float2*)(fin + (long)(c + coloff) * DFEAT);
  float2 x = src[lane];
  float* dst = fout + (long)r * DFEAT + lane * 2;
  unsafeAtomicAdd(dst, v * x.x);
  unsafeAtomicAdd(dst + 1, v * x.y);
}

// ---------------- WMMA batched dot products ----------------
// One wave handles 16 batch rows. A = 16 user rows x (K-chunk of 4),
// B = (K-chunk of 4) x 16 bundle rows; accumulate over K=64 (16 chunks),
// take the diagonal of the 16x16 f32 result. Two accumulators: pos/neg bundle.
//
// A 16x4 f32 layout: lanes 0-15 M=0-15, lanes 16-31 M=0-15;
//   VGPR0 = K{0|2}, VGPR1 = K{1|3}  -> per-lane float2 at col 4k + 2*(lane>>4).
// B 4x16 f32 layout: N = lane&15 across lanes; VGPR0 = row K{0|2}, VGPR1 = K{1|3}.
// C/D 16x16 f32: VGPR r: lanes0-15 -> M=r, lanes16-31 -> M=r+8, N=lane&15.
//   Diagonal M==N lives in lanes 0-7 (c[lane]) and lanes 24-31 (c[lane-24]).

__global__ void k_pred_wmma(const float* __restrict__ acc, const float* __restrict__ brep,
                            const int* __restrict__ users, const int* __restrict__ bundles,
                            float* __restrict__ pred) {
  int b0 = blockIdx.x * 16;
  int lane = threadIdx.x;          // 0..31, wave32
  int m = lane & 15;
  int kh = lane >> 4;              // 0 or 1 -> +0 / +2 column base

  const float* urow  = acc  + (long)users[b0 + m] * DFEAT;
  const float* brow0 = brep + (long)bundles[(b0 + m) * 2 + 0] * DFEAT;
  const float* brow1 = brep + (long)bundles[(b0 + m) * 2 + 1] * DFEAT;

  v8f c0 = {};
  v8f c1 = {};
#pragma unroll
  for (int k = 0; k < 16; ++k) {
    int off = 4 * k + 2 * kh;
    v2f a;   a.x  = urow[off];  a.y  = urow[off + 1];
    v2f bv0; bv0.x = brow0[off]; bv0.y = brow0[off + 1];
    v2f bv1; bv1.x = brow1[off]; bv1.y = brow1[off + 1];
    c0 = __builtin_amdgcn_wmma_f32_16x16x4_f32(false, a, false, bv0, (short)0, c0, false, false);
    c1 = __builtin_amdgcn_wmma_f32_16x16x4_f32(false, a, false, bv1, (short)0, c1, false, false);
  }

  // Extract diagonal elements.
  int r = -1;
  if (lane < 8) r = lane;          // M = lane   (0..7)
  else if (lane >= 24) r = lane - 24;  // M = lane-16 (8..15)
  if (r >= 0) {
    float d0 = c0[0], d1 = c1[0];
#pragma unroll
    for (int q = 1; q < 8; ++q) {
      if (r == q) { d0 = c0[q]; d1 = c1[q]; }
    }
    int row = (lane < 8) ? lane : (lane - 16);
    pred[(b0 + row) * 2 + 0] = d0;
    pred[(b0 + row) * 2 + 1] = d1;
  }
}

// ---------------- BPR loss reduction ----------------

__global__ void k_loss(const float* __restrict__ pred, float* __restrict__ out, int B) {
  __shared__ float sdata[256];
  int tid = threadIdx.x;
  float s = 0.0f;
  for (int i = tid; i < B; i += 256) {
    float x = pred[2 * i] - pred[2 * i + 1];
    // -log_sigmoid(x) = max(-x,0) + log1p(exp(-|x|))
    s += fmaxf(-x, 0.0f) + log1pf(expf(-fabsf(x)));
  }
  sdata[tid] = s;
  __syncthreads();
  for (int st = 128; st > 0; st >>= 1) {
    if (tid < st) sdata[tid] += sdata[tid + st];
    __syncthreads();
  }
  if (tid == 0) {
    out[0] = sdata[0] / (float)B;  // bpr_loss
    out[1] = 0.0f;                 // c_loss
  }
}

// ---------------- driver ----------------

extern "C" void kernel_launch(void* const* d_in, const int* in_sizes, int n_in,
                              void* d_out, int out_size, void* d_ws, size_t ws_size,
                              hipStream_t stream) {
  const float* uf        = (const float*)d_in[0];
  const float* itf       = (const float*)d_in[1];
  const float* prop_vals = (const float*)d_in[2];
  const float* bi_vals   = (const float*)d_in[3];
  const int*   prop_rows = (const int*)d_in[4];
  const int*   prop_cols = (const int*)d_in[5];
  const int*   bi_rows   = (const int*)d_in[6];
  const int*   bi_cols   = (const int*)d_in[7];
  const int*   users     = (const int*)d_in[8];
  const int*   bundles   = (const int*)d_in[9];
  float* out = (float*)d_out;

  const long nu_elems = in_sizes[0];           // NU*64
  const long ni_elems = in_sizes[1];           // NI*64
  const long n_elems  = nu_elems + ni_elems;   // N*64
  const int  nnz_prop = in_sizes[2];
  const int  nnz_bi   = in_sizes[3];
  const int  NU       = (int)(nu_elems / DFEAT);
  const int  B        = in_sizes[8];
  const int  NB       = 20000;

  float* acc  = (float*)d_ws;
  float* f0   = acc + n_elems;
  float* f1   = f0 + n_elems;
  float* brep = f1 + n_elems;
  float* pred = brep + (long)NB * DFEAT;

  const int T = 256;
  const long nb_elem = (n_elems + T - 1) / T;
  const long spmm_blocks    = ((long)nnz_prop * 32 + T - 1) / T;
  const long spmm_bi_blocks = ((long)nnz_bi   * 32 + T - 1) / T;
  const long brep_elems = (long)NB * DFEAT;

  // acc = f0 = concat(users, items); f1 = 0
  k_init<<<(unsigned)nb_elem, T, 0, stream>>>(uf, itf, acc, f0, f1, n_elems, nu_elems);
  // layer 1: f1 = A f0 ; acc += f1 ; f0 = 0
  k_spmm<<<(unsigned)spmm_blocks, T, 0, stream>>>(prop_rows, prop_cols, prop_vals, f0, f1, nnz_prop, 0);
  k_acc_add_zero<<<(unsigned)nb_elem, T, 0, stream>>>(acc, f1, f0, n_elems);
  // layer 2: f0 = A f1 ; acc = (acc + f0) / 3
  k_spmm<<<(unsigned)spmm_blocks, T, 0, stream>>>(prop_rows, prop_cols, prop_vals, f1, f0, nnz_prop, 0);
  k_acc_add_scale<<<(unsigned)nb_elem, T, 0, stream>>>(acc, f0, n_elems);
  // bundles_rep = B_bi * items_rep  (items live at row offset NU in acc)
  k_zero<<<(unsigned)((brep_elems + T - 1) / T), T, 0, stream>>>(brep, brep_elems);
  k_spmm<<<(unsigned)spmm_bi_blocks, T, 0, stream>>>(bi_rows, bi_cols, bi_vals, acc, brep, nnz_bi, NU);
  // predictions via WMMA (16 batch rows / wave), then loss
  k_pred_wmma<<<B / 16, 32, 0, stream>>>(acc, brep, users, bundles, pred);
  k_loss<<<1, 256, 0, stream>>>(pred, out, B);
}